// HungarianMatcher_60129542923
// MI455X (gfx1250) — compile-verified
//
#include <hip/hip_runtime.h>
#include <hip/hip_bf16.h>

// ---------------------------------------------------------------------------
// DETR Hungarian cost matrix on MI455X (gfx1250).
//   C[n,m] = 5*L1(pred_box[n], tgt_box[m]) + 2*focal_class(n, tgt_id[m])
//            - 2*GIoU(pred_box[n], tgt_box[m])
// Focal class cost depends only on (n, class) -> precompute a transposed
// [nc, N] table once (4.1M exp/log), then the main kernel is pure VALU +
// coalesced 102MB store stream. Tiles staged with CDNA5 async global->LDS.
// ---------------------------------------------------------------------------

#define AS1 __attribute__((address_space(1)))
#define AS3 __attribute__((address_space(3)))

typedef int v4i __attribute__((vector_size(16)));

#define NT 32    // pred rows per block tile
#define MT 256   // target columns per block tile (== blockDim.x)

#if __has_builtin(__builtin_amdgcn_global_load_async_to_lds_b128) && \
    __has_builtin(__builtin_amdgcn_global_load_async_to_lds_b32)
#define USE_ASYNC_LDS 1
#else
#define USE_ASYNC_LDS 0
#endif

__device__ __forceinline__ float class_cost_from_logit(float x) {
  // p = sigmoid(x); pos = a*(1-p)^2*(-log(p+eps)); neg = (1-a)*p^2*(-log(1-p+eps))
  float p   = 1.0f / (1.0f + __expf(-x));
  float omp = 1.0f - p;
  float pos = 0.25f * omp * omp * (-__logf(p + 1e-8f));
  float neg = 0.75f * p * p * (-__logf(omp + 1e-8f));
  return pos - neg;
}

__global__ __launch_bounds__(256) void class_table_kernel(
    const float* __restrict__ logits, float* __restrict__ Tt, int N, int nc) {
  int idx = blockIdx.x * 256 + threadIdx.x;
  if (idx >= N * nc) return;
  int n = idx / nc;
  int c = idx - n * nc;
  // transposed table: row per class, consecutive in n (so the cost kernel
  // reads 32 consecutive floats per column with float4 loads)
  Tt[(size_t)c * N + n] = class_cost_from_logit(logits[idx]);
}

template <bool USE_TABLE>
__global__ __launch_bounds__(256) void cost_kernel(
    const float* __restrict__ pred_boxes,   // [N,4] cxcywh
    const float* __restrict__ tgt_boxes,    // [M,4] cxcywh
    const int*   __restrict__ tgt_labels,   // [M]
    const float* __restrict__ cls_src,      // USE_TABLE ? Tt[nc,N] : logits[N,nc]
    float* __restrict__ C,                  // [N,M]
    int N, int M, int nc) {
  __shared__ float sTgtRaw[MT * 4];   // raw target boxes for this tile
  __shared__ float spRaw[NT * 4];     // raw pred boxes for this tile
  __shared__ float sp[NT][9];         // cx,cy,w,h,x0,y0,x1,y1,area

  const int tid   = threadIdx.x;
  const int n0    = blockIdx.y * NT;
  const int m     = blockIdx.x * MT + tid;
  const bool valid = (m < M);

  // ---- stage tiles into LDS (CDNA5 async global->LDS when available) ----
#if USE_ASYNC_LDS
  if (valid) {
    __builtin_amdgcn_global_load_async_to_lds_b128(
        (AS1 v4i*)(v4i*)(void*)const_cast<float*>(tgt_boxes + (size_t)m * 4),
        (AS3 v4i*)(v4i*)(void*)&sTgtRaw[tid * 4], 0, 0);
  }
  if (tid < NT * 4 && (n0 * 4 + tid) < N * 4) {
    __builtin_amdgcn_global_load_async_to_lds_b32(
        (AS1 int*)(int*)(void*)const_cast<float*>(pred_boxes + (size_t)n0 * 4 + tid),
        (AS3 int*)(int*)(void*)&spRaw[tid], 0, 0);
  }
#if __has_builtin(__builtin_amdgcn_s_wait_asynccnt)
  __builtin_amdgcn_s_wait_asynccnt(0);
#else
  asm volatile("s_wait_asynccnt 0" ::: "memory");
#endif
#else
  if (valid) {
    float4 tb = ((const float4*)tgt_boxes)[m];
    *(float4*)&sTgtRaw[tid * 4] = tb;
  }
  if (tid < NT * 4) {
    int idx = n0 * 4 + tid;
    spRaw[tid] = (idx < N * 4) ? pred_boxes[idx] : 0.0f;
  }
#endif
  __syncthreads();

  // per-row derived pred quantities
  if (tid < NT) {
    float cx = spRaw[tid * 4 + 0], cy = spRaw[tid * 4 + 1];
    float w  = spRaw[tid * 4 + 2], h  = spRaw[tid * 4 + 3];
    float hw = 0.5f * w, hh = 0.5f * h;
    float x0 = cx - hw, y0 = cy - hh, x1 = cx + hw, y1 = cy + hh;
    sp[tid][0] = cx; sp[tid][1] = cy; sp[tid][2] = w;  sp[tid][3] = h;
    sp[tid][4] = x0; sp[tid][5] = y0; sp[tid][6] = x1; sp[tid][7] = y1;
    sp[tid][8] = (x1 - x0) * (y1 - y0);
  }

  // per-thread target (one column per thread, fixed for whole tile)
  float tcx = 0.f, tcy = 0.f, tw = 0.f, th = 0.f;
  float tx0 = 0.f, ty0 = 0.f, tx1 = 0.f, ty1 = 0.f, tarea = 0.f;
  int id = 0;
  if (valid) {
    tcx = sTgtRaw[tid * 4 + 0]; tcy = sTgtRaw[tid * 4 + 1];
    tw  = sTgtRaw[tid * 4 + 2]; th  = sTgtRaw[tid * 4 + 3];
    float hw = 0.5f * tw, hh = 0.5f * th;
    tx0 = tcx - hw; ty0 = tcy - hh; tx1 = tcx + hw; ty1 = tcy + hh;
    tarea = (tx1 - tx0) * (ty1 - ty0);
    id = tgt_labels[m];
    id = id < 0 ? 0 : (id > nc - 1 ? nc - 1 : id);
  }
  __syncthreads();

  if (!valid) return;

  // class costs for all 32 rows of this tile: 8x float4 from transposed table
  float cls[NT];
  if (USE_TABLE) {
    const float* base = cls_src + (size_t)id * N + n0;
    if ((n0 + NT <= N) && ((N & 3) == 0)) {
#pragma unroll
      for (int k = 0; k < NT / 4; ++k) {
        float4 v = ((const float4*)base)[k];
        cls[4 * k + 0] = v.x; cls[4 * k + 1] = v.y;
        cls[4 * k + 2] = v.z; cls[4 * k + 3] = v.w;
      }
    } else {
#pragma unroll
      for (int i = 0; i < NT; ++i)
        cls[i] = (n0 + i < N) ? base[i] : 0.0f;
    }
  }

  int rows = N - n0;
  if (rows > NT) rows = NT;
  float* out = C + (size_t)n0 * M + m;

#pragma unroll 4
  for (int i = 0; i < rows; ++i) {
    float pcx = sp[i][0], pcy = sp[i][1], pw = sp[i][2], ph = sp[i][3];
    float px0 = sp[i][4], py0 = sp[i][5], px1 = sp[i][6], py1 = sp[i][7];
    float parea = sp[i][8];

    // L1 on cxcywh
    float l1 = fabsf(pcx - tcx) + fabsf(pcy - tcy) +
               fabsf(pw - tw) + fabsf(ph - th);

    // GIoU on xyxy
    float ix0 = fmaxf(px0, tx0), iy0 = fmaxf(py0, ty0);
    float ix1 = fminf(px1, tx1), iy1 = fminf(py1, ty1);
    float iw = fmaxf(ix1 - ix0, 0.0f), ih = fmaxf(iy1 - iy0, 0.0f);
    float inter = iw * ih;
    float uni = parea + tarea - inter;
    float cx0 = fminf(px0, tx0), cy0 = fminf(py0, ty0);
    float cx1 = fmaxf(px1, tx1), cy1 = fmaxf(py1, ty1);
    float cw = fmaxf(cx1 - cx0, 0.0f), ch = fmaxf(cy1 - cy0, 0.0f);
    float carea = cw * ch;
    float giou = inter / uni - (carea - uni) / carea;

    float ccls = USE_TABLE
                     ? cls[i]
                     : class_cost_from_logit(cls_src[(size_t)(n0 + i) * nc + id]);

    out[(size_t)i * M] = 5.0f * l1 + 2.0f * ccls - 2.0f * giou;
  }
}

extern "C" void kernel_launch(void* const* d_in, const int* in_sizes, int n_in,
                              void* d_out, int out_size, void* d_ws, size_t ws_size,
                              hipStream_t stream) {
  const float* logits = (const float*)d_in[0];   // [16,1000,256] f32
  const float* pboxes = (const float*)d_in[1];   // [16,1000,4]  f32
  const int*   labels = (const int*)d_in[2];     // [1600]       int
  const float* tboxes = (const float*)d_in[3];   // [1600,4]     f32
  float* out = (float*)d_out;

  const int N  = in_sizes[1] / 4;                // 16000 pred boxes
  const int M  = in_sizes[2];                    // 1600 targets
  const int nc = (N > 0) ? in_sizes[0] / N : 1;  // 256 classes

  dim3 gridB((M + MT - 1) / MT, (N + NT - 1) / NT);
  const size_t tbl_bytes = (size_t)N * (size_t)nc * sizeof(float);

  if (tbl_bytes <= ws_size) {
    float* Tt = (float*)d_ws;
    int total = N * nc;
    class_table_kernel<<<(total + 255) / 256, 256, 0, stream>>>(logits, Tt, N, nc);
    cost_kernel<true><<<gridB, 256, 0, stream>>>(pboxes, tboxes, labels, Tt,
                                                 out, N, M, nc);
  } else {
    // scratch too small: fuse class cost inline (more transcendentals, same result)
    cost_kernel<false><<<gridB, 256, 0, stream>>>(pboxes, tboxes, labels, logits,
                                                  out, N, M, nc);
  }
}